// NodeSpanEncoder_31129922962206
// MI455X (gfx1250) — compile-verified
//
#include <hip/hip_runtime.h>
#include <stdint.h>

// Problem constants (fixed by the reference).
constexpr int B = 16, T = 4096, N = 1024, D = 128;

constexpr int DTILE   = 16;            // D-columns per block
constexpr int NT      = D / DTILE;     // 8 tiles -> grid = B*NT = 128 blocks
constexpr int THREADS = 512;           // 16 wave32s per block
constexpr int CHUNK   = 128;           // rows per scan chunk
constexpr int NCHUNK  = T / CHUNK;     // 32 chunks (32*16 = 512 = THREADS)

// Padded LDS panel: word offset of (t, col) = t*16 + (t>>2)*4 + col.
// -> every row base is 16-B aligned (b128 async OK) and 16 consecutive
//    cols map to 16 distinct banks (pad of 4 dwords per 4 rows).
__device__ __forceinline__ int paddr(int t, int c) {
  return (t << 4) + ((t >> 2) << 2) + c;
}

constexpr uint32_t P_BYTES  = (uint32_t)(T / 4) * 68u * 4u;   // 278528 B
constexpr uint32_t S_OFF    = P_BYTES;                        // starts: 4 KB
constexpr uint32_t E_OFF    = S_OFF + (uint32_t)N * 4u;       // ends:   4 KB
constexpr uint32_t TOT_OFF  = E_OFF + (uint32_t)N * 4u;       // chunk totals
constexpr uint32_t LDS_BYTES = TOT_OFF + (uint32_t)NCHUNK * DTILE * 4u; // 288768 B (< 320 KB)

__device__ __forceinline__ void async_ld_b128(uint32_t lds_off, const void* g) {
  // GLOBAL_LOAD_ASYNC_TO_LDS_B128: per-lane LDS dest offset in a VGPR,
  // 64-bit global address in a VGPR pair, tracked by ASYNCcnt.
  asm volatile("global_load_async_to_lds_b128 %0, %1, off"
               :: "v"(lds_off), "v"(g) : "memory");
}

__global__ __launch_bounds__(THREADS) void span_mean_kernel(
    const float* __restrict__ x, const int* __restrict__ starts,
    const int* __restrict__ ends, float* __restrict__ out) {
  extern __shared__ unsigned char smem[];
  float* P   = (float*)smem;
  int*   S   = (int*)(smem + S_OFF);
  int*   E   = (int*)(smem + E_OFF);
  float* TOT = (float*)(smem + TOT_OFF);

  const int tid = threadIdx.x;
  const int b   = blockIdx.x / NT;
  const int d0  = (blockIdx.x % NT) * DTILE;

  // ---- Phase 0: async-copy x[b, :, d0:d0+16] + span indices into LDS ----
  const float* xb = x + ((size_t)b * T) * D + d0;
  for (int k = 0; k < (T * DTILE / 4) / THREADS; ++k) {   // 32 iters
    int idx = tid + k * THREADS;      // 0..16383 ; 4 lanes cover one row's 64 B
    int t = idx >> 2;
    int q = idx & 3;                  // which float4 of the 16-col row
    uint32_t lds_off = (uint32_t)(paddr(t, q << 2)) * 4u; // 16-B aligned
    async_ld_b128(lds_off, xb + (size_t)t * D + (q << 2));
  }
  {
    // 1024 starts + 1024 ends = 512 b128 transfers, one per thread.
    const int* g = (tid < 256) ? (starts + (size_t)b * N) : (ends + (size_t)b * N);
    uint32_t base = (tid < 256) ? S_OFF : E_OFF;
    int tt = tid & 255;
    async_ld_b128(base + (uint32_t)tt * 16u, g + tt * 4);
  }
  asm volatile("s_wait_asynccnt 0" ::: "memory");
  __syncthreads();

  // ---- Phase 1: inclusive prefix sum along T (two-level scan in LDS) ----
  const int chunk = tid >> 4;   // 0..31
  const int col   = tid & 15;   // 0..15
  {
    float run = 0.f;
    const int base = chunk * CHUNK;
    for (int r = 0; r < CHUNK; ++r) {
      int a = paddr(base + r, col);
      run += P[a];
      P[a] = run;
    }
    TOT[chunk * DTILE + col] = run;
  }
  __syncthreads();
  {
    float off = 0.f;
    for (int c = 0; c < chunk; ++c) off += TOT[c * DTILE + col];
    if (chunk != 0) {
      const int base = chunk * CHUNK;
      for (int r = 0; r < CHUNK; ++r) P[paddr(base + r, col)] += off;
    }
  }
  __syncthreads();

  // ---- Phase 2: gather span means, coalesced float4 stores ----
  float* ob = out + ((size_t)b * N) * D + d0;
  for (int k = 0; k < (N * DTILE / 4) / THREADS; ++k) {   // 8 iters
    int i = tid + k * THREADS;   // 0..4095
    int n = i >> 2;
    int q = (i & 3) << 2;        // starting col of this float4
    int s = S[n], e = E[n];
    float cnt = (float)(e - s + 1);
    float lo0 = 0.f, lo1 = 0.f, lo2 = 0.f, lo3 = 0.f;
    if (s > 0) {
      int la = paddr(s - 1, q);
      lo0 = P[la]; lo1 = P[la + 1]; lo2 = P[la + 2]; lo3 = P[la + 3];
    }
    int ha = paddr(e, q);
    float4 v;
    v.x = (P[ha]     - lo0) / cnt;
    v.y = (P[ha + 1] - lo1) / cnt;
    v.z = (P[ha + 2] - lo2) / cnt;
    v.w = (P[ha + 3] - lo3) / cnt;
    *(float4*)(ob + (size_t)n * D + q) = v;
  }
}

extern "C" void kernel_launch(void* const* d_in, const int* in_sizes, int n_in,
                              void* d_out, int out_size, void* d_ws, size_t ws_size,
                              hipStream_t stream) {
  (void)in_sizes; (void)n_in; (void)d_ws; (void)ws_size; (void)out_size;
  const float* x      = (const float*)d_in[0];
  const int*   starts = (const int*)d_in[1];
  const int*   ends   = (const int*)d_in[2];
  float*       out    = (float*)d_out;

  span_mean_kernel<<<dim3(B * NT), dim3(THREADS), LDS_BYTES, stream>>>(
      x, starts, ends, out);
}